// ExponentialMovingAverage_4818953306240
// MI455X (gfx1250) — compile-verified
//
#include <hip/hip_runtime.h>

// Problem constants (match reference)
#define BB 32
#define CC 128
#define TT 8192
#define NSEG 64
#define SEGLEN 128          // TT / NSEG
#define TILES_PER_SEG 8     // SEGLEN / 16
#define BTILE 16
#define CTILE 32
#define TSTRIDE 18          // LDS t-stride (floats)
#define CSTRIDE 289         // LDS c-stride = 16*18+1 (floats), odd -> conflict-free c-major reads

typedef float v2f __attribute__((ext_vector_type(2)));
typedef float v8f __attribute__((ext_vector_type(8)));

__device__ __forceinline__ float clamp01(float v) {
    return fminf(fmaxf(v, 0.0f), 1.0f);
}

// branchless integer power, e in [0,15]
__device__ __forceinline__ float powi(float a, int e) {
    float r = 1.0f, p = a;
    #pragma unroll
    for (int i = 0; i < 4; ++i) {
        r = (e & 1) ? r * p : r;
        p *= p;
        e >>= 1;
    }
    return r;
}

// ---------------------------------------------------------------------------
// Pass A: per-(b,c,segment) reduction with zero init (segment 0 seeded by x0).
// yseg[(b*CC+c)*NSEG + s] = last value of segment-local recurrence.
// ---------------------------------------------------------------------------
__global__ __launch_bounds__(256) void ema_seg_reduce(
    const float* __restrict__ x, const float* __restrict__ wts,
    float* __restrict__ yseg)
{
    int tid = blockIdx.x * blockDim.x + threadIdx.x;
    if (tid >= BB * CC * NSEG) return;
    int s = tid % NSEG;
    int c = (tid / NSEG) % CC;
    // b = tid / (NSEG*CC) implicit in flat offset below
    float w  = clamp01(wts[c]);
    float al = 1.0f - w;

    size_t bc = (size_t)(tid / NSEG);              // b*CC + c
    const float4* p = (const float4*)(x + bc * TT + (size_t)s * SEGLEN);

    float4 v = p[0];
    float a = (s == 0) ? v.x : (w * v.x);          // zero-init: a = w*x0; seg0: seed
    a = al * a + w * v.y;
    a = al * a + w * v.z;
    a = al * a + w * v.w;
    #pragma unroll 4
    for (int q = 1; q < SEGLEN / 4; ++q) {
        v = p[q];
        a = al * a + w * v.x;
        a = al * a + w * v.y;
        a = al * a + w * v.z;
        a = al * a + w * v.w;
    }
    yseg[tid] = a;
}

// ---------------------------------------------------------------------------
// Pass B: per-(b,c) scan over segment carries: carry_s = y0_s + al^SEGLEN*carry_{s-1}
// ---------------------------------------------------------------------------
__global__ __launch_bounds__(256) void ema_seg_scan(
    const float* __restrict__ wts, const float* __restrict__ yseg,
    float* __restrict__ carryOut)
{
    int tid = blockIdx.x * blockDim.x + threadIdx.x;
    if (tid >= BB * CC) return;
    int c = tid % CC;
    float w  = clamp01(wts[c]);
    float al = 1.0f - w;
    float alL = al;
    #pragma unroll
    for (int i = 0; i < 7; ++i) alL *= alL;        // al^128

    const float* ys = yseg + (size_t)tid * NSEG;
    float* co = carryOut + (size_t)tid * NSEG;
    float carry = ys[0];                           // segment 0 already seeded
    co[0] = carry;
    for (int s = 1; s < NSEG; ++s) {
        carry = ys[s] + alL * carry;
        co[s] = carry;
    }
}

// ---------------------------------------------------------------------------
// Pass C: WMMA blocked scan. Workgroup = (b-tile of 16, c-tile of 32, segment).
// 32 waves, one channel each. Per 16-step tile:
//   D[t,n] = sum_j M[t,j]*X[j,n] + al^(t+1)*carry[n]   via 4x wmma_f32_16x16x4_f32
// D tiles staged in LDS, stored as coalesced [b,t,c] lines.
// ---------------------------------------------------------------------------
__global__ __launch_bounds__(1024) void ema_scan_wmma(
    const float* __restrict__ x, const float* __restrict__ wts,
    const float* __restrict__ carryIn, float* __restrict__ out)
{
    __shared__ float lds[CTILE * CSTRIDE];         // ~36 KB

    const int seg   = blockIdx.x % NSEG;
    const int ctile = (blockIdx.x / NSEG) % (CC / CTILE);
    const int btile =  blockIdx.x / (NSEG * (CC / CTILE));
    const int b0 = btile * BTILE;
    const int c0 = ctile * CTILE;

    const int lane = threadIdx.x & 31;
    const int wave = threadIdx.x >> 5;
    const int n    = lane & 15;                    // column (batch) / row index m
    const int half = lane >> 4;

    const int c = c0 + wave;
    const float w  = clamp01(wts[c]);
    const float al = 1.0f - w;

    // Build A chunks: A[kc].{x,y} = M[m][4kc+2h], M[m][4kc+2h+1], M[t][j]=w*al^(t-j), lower-tri
    v2f A[4];
    #pragma unroll
    for (int kc = 0; kc < 4; ++kc) {
        int j0 = 4 * kc + 2 * half;
        A[kc].x = (j0     <= n) ? w * powi(al, n - j0)     : 0.0f;
        A[kc].y = (j0 + 1 <= n) ? w * powi(al, n - j0 - 1) : 0.0f;
    }
    // Special A chunk 0 for the very first tile (t=0 seeded by x0: column 0 coeff al^t not w*al^t)
    v2f A0first = A[0];
    if (half == 0) A0first.x = powi(al, n);

    // Incoming carry per batch column
    float carry = 0.0f;
    if (seg > 0)
        carry = carryIn[((size_t)(b0 + n) * CC + c) * NSEG + (seg - 1)];

    const float sbase = powi(al, 8 * half + 1);    // al^(m+1) base for this half's D rows
    const float* xb = x + ((size_t)(b0 + n) * CC + c) * TT + (size_t)seg * SEGLEN + 2 * half;

    for (int tile = 0; tile < TILES_PER_SEG; ++tile) {
        const int tg = seg * SEGLEN + tile * 16;

        // Load B: lane holds X[t][n] for its K rows; two consecutive t per chunk -> 8B loads
        v2f Bm[4];
        #pragma unroll
        for (int kc = 0; kc < 4; ++kc)
            Bm[kc] = *(const v2f*)(xb + tile * 16 + 4 * kc);

        // C accumulator = outer(al^(m+1), carry)
        v8f d;
        float s = sbase;
        #pragma unroll
        for (int r = 0; r < 8; ++r) { d[r] = s * carry; s *= al; }

        v2f a0 = (seg == 0 && tile == 0) ? A0first : A[0];
        d = __builtin_amdgcn_wmma_f32_16x16x4_f32(false, a0,   false, Bm[0], (short)0, d, false, false);
        d = __builtin_amdgcn_wmma_f32_16x16x4_f32(false, A[1], false, Bm[1], (short)0, d, false, false);
        d = __builtin_amdgcn_wmma_f32_16x16x4_f32(false, A[2], false, Bm[2], (short)0, d, false, false);
        d = __builtin_amdgcn_wmma_f32_16x16x4_f32(false, A[3], false, Bm[3], (short)0, d, false, false);

        // carry for next tile: D row 15 (VGPR 7, lanes 16..31)
        carry = __shfl(d[7], 16 + n, 32);

        // Stage D tile into LDS: [c_local][t][b], conflict-free strides
        const int base = wave * CSTRIDE;
        #pragma unroll
        for (int r = 0; r < 8; ++r)
            lds[base + (r + 8 * half) * TSTRIDE + n] = d[r];

        __syncthreads();

        // Cooperative coalesced store: lanes sweep c -> contiguous 128B lines of out[b,t,c]
        {
            const int cc = threadIdx.x & 31;
            const int tt = (threadIdx.x >> 5) & 15;
            const int bh = threadIdx.x >> 9;
            #pragma unroll
            for (int k = 0; k < 8; ++k) {
                const int bb = bh * 8 + k;
                out[((size_t)(b0 + bb) * TT + tg + tt) * CC + c0 + cc] =
                    lds[cc * CSTRIDE + tt * TSTRIDE + bb];
            }
        }
        __syncthreads();
    }
}

extern "C" void kernel_launch(void* const* d_in, const int* in_sizes, int n_in,
                              void* d_out, int out_size, void* d_ws, size_t ws_size,
                              hipStream_t stream) {
    const float* x   = (const float*)d_in[0];
    const float* wts = (const float*)d_in[1];
    float* out  = (float*)d_out;

    float* yseg  = (float*)d_ws;                       // BB*CC*NSEG floats = 1 MiB
    float* carry = yseg + (size_t)BB * CC * NSEG;      // BB*CC*NSEG floats = 1 MiB

    ema_seg_reduce<<<(BB * CC * NSEG + 255) / 256, 256, 0, stream>>>(x, wts, yseg);
    ema_seg_scan<<<(BB * CC + 255) / 256, 256, 0, stream>>>(wts, yseg, carry);
    ema_scan_wmma<<<(BB / BTILE) * (CC / CTILE) * NSEG, 1024, 0, stream>>>(x, wts, carry, out);
}